// RM_UNet_82815559401678
// MI455X (gfx1250) — compile-verified
//
#include <hip/hip_runtime.h>
#include <hip/hip_bf16.h>
#include <math.h>

// ---------------------------------------------------------------------------
// Types for CDNA5 fp32 WMMA (V_WMMA_F32_16X16X4_F32):
//   A: 16x4 f32  -> 2 VGPRs/lane, B: 4x16 f32 -> 2 VGPRs/lane, C/D: 8 VGPRs
// ---------------------------------------------------------------------------
typedef __attribute__((ext_vector_type(2))) float v2f;
typedef __attribute__((ext_vector_type(8))) float v8f;

// address-space-qualified int for the async-to-LDS builtin's prototype
typedef __attribute__((address_space(1))) int as1_int;   // global
typedef __attribute__((address_space(3))) int as3_int;   // LDS

__device__ __forceinline__ float sigmoidf_(float x) { return 1.f / (1.f + __expf(-x)); }
__device__ __forceinline__ float softplusf_(float x) { return x > 20.f ? x : log1pf(__expf(x)); }
__device__ __forceinline__ float eluf_(float x) { return x > 0.f ? x : (__expf(x) - 1.f); }

// ---------------------------------------------------------------------------
// Generic fp32 WMMA GEMM:  D[M,N] = A[M,K] * W[N,K]^T   (one wave = 16x16 tile)
// act: 0 = none, 1 = softplus(v + bias[col])
// ---------------------------------------------------------------------------
__global__ void k_gemm_wmma_f32(const float* __restrict__ A, const float* __restrict__ W,
                                float* __restrict__ D, int M, int N, int K,
                                int lda, int ldb, int ldd,
                                const float* __restrict__ bias, int act) {
    const int wavesPerBlock = blockDim.x >> 5;
    int wave = blockIdx.x * wavesPerBlock + (threadIdx.x >> 5);
    int tilesN = N >> 4;
    int tiles = (M >> 4) * tilesN;
    if (wave >= tiles) return;            // uniform per-wave: EXEC stays all-1s for WMMA
    int tm = (wave / tilesN) << 4;
    int tn = (wave % tilesN) << 4;
    int lane = threadIdx.x & 31;
    int r  = lane & 15;                   // M (for A) / N (for B) within tile
    int kh = (lane >> 4) << 1;            // lanes 16-31 hold K=2,3 of each k-quad

    const float* Ap = A + (size_t)(tm + r) * lda + kh;
    const float* Wp = W + (size_t)(tn + r) * ldb + kh;

    v8f acc = {};
    for (int k = 0; k < K; k += 4) {
        v2f av, bv;
        av.x = Ap[k];     av.y = Ap[k + 1];
        bv.x = Wp[k];     bv.y = Wp[k + 1];
        acc = __builtin_amdgcn_wmma_f32_16x16x4_f32(false, av, false, bv,
                                                    (short)0, acc, false, false);
    }

    int col = tn + (lane & 15);
    int mh  = (lane >> 4) << 3;           // D rows: vgpr i -> M = i (+8 for lanes 16-31)
    float badd = bias ? bias[col] : 0.f;
#pragma unroll
    for (int i = 0; i < 8; ++i) {
        float v = acc[i];
        if (act == 1) v = softplusf_(v + badd);
        D[(size_t)(tm + mh + i) * ldd + col] = v;
    }
}

// ---------------------------------------------------------------------------
// Encoder: Conv1d(stride 2, pad 6, K=13) + GLU  (out channels = OCh after GLU)
// ---------------------------------------------------------------------------
__global__ void k_conv_glu(const float* __restrict__ in, const float* __restrict__ w,
                           const float* __restrict__ bias, float* __restrict__ out,
                           int B, int IC, int OCh, int Lin, int Lout) {
    int idx = blockIdx.x * blockDim.x + threadIdx.x;
    int total = B * OCh * Lout;
    if (idx >= total) return;
    int lo = idx % Lout;
    int co = (idx / Lout) % OCh;
    int b  = idx / (Lout * OCh);
    __builtin_prefetch(w + ((size_t)co * IC) * 13, 0, 0);          // global_prefetch_b8
    __builtin_prefetch(w + ((size_t)(co + OCh) * IC) * 13, 0, 0);
    float yo = bias[co];
    float yg = bias[co + OCh];
    int base = lo * 2 - 6;
    for (int k = 0; k < 13; ++k) {
        int l = base + k;
        if (l < 0 || l >= Lin) continue;
        const float* ip = in + ((size_t)b * IC) * Lin + l;
        const float* wo = w + ((size_t)co * IC) * 13 + k;
        const float* wg = w + ((size_t)(co + OCh) * IC) * 13 + k;
        for (int ic = 0; ic < IC; ++ic) {
            float xv = ip[(size_t)ic * Lin];
            yo += xv * wo[(size_t)ic * 13];
            yg += xv * wg[(size_t)ic * 13];
        }
    }
    out[idx] = yo * sigmoidf_(yg);
}

// ---------------------------------------------------------------------------
// Decoder: ConvTranspose1d(K=13) [+ ELU + skip]   mode 1 = elu+skip, 0 = plain
// w layout: (IC, OC, 13)
// ---------------------------------------------------------------------------
__global__ void k_convT(const float* __restrict__ in, const float* __restrict__ w,
                        const float* __restrict__ bias, const float* __restrict__ skip,
                        float* __restrict__ out, int B, int IC, int OC,
                        int Lin, int Lout, int stride, int mode) {
    int idx = blockIdx.x * blockDim.x + threadIdx.x;
    int total = B * OC * Lout;
    if (idx >= total) return;
    int t  = idx % Lout;
    int oc = (idx / Lout) % OC;
    int b  = idx / (Lout * OC);
    __builtin_prefetch(w + (size_t)oc * 13, 0, 0);                 // global_prefetch_b8
    float acc = bias[oc];
    for (int k = 0; k < 13; ++k) {
        int num = t + 6 - k;
        if (num < 0) continue;
        if (num % stride) continue;
        int l = num / stride;
        if (l >= Lin) continue;
        const float* ip = in + ((size_t)b * IC) * Lin + l;
        const float* wp = w + (size_t)oc * 13 + k;
        for (int ic = 0; ic < IC; ++ic)
            acc += ip[(size_t)ic * Lin] * wp[(size_t)ic * OC * 13];
    }
    if (mode) out[idx] = eluf_(acc) + skip[idx];
    else      out[idx] = acc;
}

// ---------------------------------------------------------------------------
// BatchNorm1d (training stats over batch+length): per-channel reduce -> scale/shift
// ---------------------------------------------------------------------------
__global__ void k_bn_stats(const float* __restrict__ x, const float* __restrict__ g,
                           const float* __restrict__ be, float* __restrict__ stats,
                           int B, int C, int L) {
    __shared__ float s1[256];
    __shared__ float s2[256];
    int c = blockIdx.x;
    int tid = threadIdx.x;
    float sum = 0.f, sq = 0.f;
    int n = B * L;
    for (int i = tid; i < n; i += 256) {
        int b = i / L, p = i - b * L;
        float v = x[((size_t)b * C + c) * L + p];
        sum += v; sq += v * v;
    }
    s1[tid] = sum; s2[tid] = sq;
    __syncthreads();
    for (int off = 128; off > 0; off >>= 1) {
        if (tid < off) { s1[tid] += s1[tid + off]; s2[tid] += s2[tid + off]; }
        __syncthreads();
    }
    if (tid == 0) {
        float inv = 1.f / (float)n;
        float m = s1[0] * inv;
        float var = s2[0] * inv - m * m;
        float sc = g[c] * rsqrtf(var + 1e-5f);
        stats[2 * c]     = sc;
        stats[2 * c + 1] = be[c] - m * sc;
    }
}

__global__ void k_bn_apply(const float* __restrict__ x, const float* __restrict__ stats,
                           float* __restrict__ y, int C, int L, int total) {
    int i = blockIdx.x * blockDim.x + threadIdx.x;
    if (i >= total) return;
    int c = (i / L) % C;
    y[i] = x[i] * stats[2 * c] + stats[2 * c + 1];
}

// ---------------------------------------------------------------------------
// Mamba pieces
// ---------------------------------------------------------------------------
__global__ void k_prep(const float* __restrict__ Alog, const float* __restrict__ wx,
                       float* __restrict__ Abuf, float* __restrict__ wxp) {
    int i = blockIdx.x * blockDim.x + threadIdx.x;
    if (i < 128 * 64) Abuf[i] = -__expf(Alog[i]);
    if (i < 144 * 128) {
        int n = i / 128, k = i - n * 128;
        wxp[i] = (n < 132) ? wx[n * 128 + k] : 0.f;
    }
}

// depthwise causal conv (K=3, left pad 2) + SiLU;  xi = first 128 cols of xz
__global__ void k_dwconv_silu(const float* __restrict__ xz, const float* __restrict__ cw,
                              const float* __restrict__ cb, float* __restrict__ u, int B) {
    int i = blockIdx.x * blockDim.x + threadIdx.x;
    int total = B * 64 * 128;
    if (i >= total) return;
    int d = i & 127;
    int l = (i >> 7) & 63;
    int b = i >> 13;
    float acc = cb[d];
#pragma unroll
    for (int j = 0; j < 3; ++j) {
        int ll = l - 2 + j;
        if (ll >= 0) acc += xz[((size_t)b * 64 + ll) * 256 + d] * cw[d * 3 + j];
    }
    u[i] = acc * sigmoidf_(acc);
}

// ---------------------------------------------------------------------------
// Selective scan: one block per batch, thread = d_inner channel, h[64] in regs.
// The per-step B/C row (128 floats) is staged into LDS with a double-buffered
// GLOBAL_LOAD_ASYNC_TO_LDS pipeline: prefetch step l+1 while computing step l,
// synchronized with s_wait_asynccnt (async loads complete in order).
// ---------------------------------------------------------------------------
#if __has_builtin(__builtin_amdgcn_global_load_async_to_lds_b32)
#define SCAN_ASYNC 1
#else
#define SCAN_ASYNC 0
#endif

__global__ void __launch_bounds__(128)
k_mamba_scan(const float* __restrict__ dt, const float* __restrict__ u,
             const float* __restrict__ xdbl, const float* __restrict__ Abuf,
             const float* __restrict__ Dp, float* __restrict__ y) {
    int b = blockIdx.x;
    int d = threadIdx.x;
    __shared__ float sbuf[2][128];   // [parity][0:64)=B row, [64:128)=C row
    float h[64];
    float Ar[64];
#pragma unroll
    for (int s = 0; s < 64; ++s) { h[s] = 0.f; Ar[s] = Abuf[d * 64 + s]; }
    float Dd = Dp[d];

#if SCAN_ASYNC
    {   // prime the pipeline: stage row l=0 into parity 0
        const float* row0 = xdbl + (size_t)b * 64 * 144;
        const float* g0 = (d < 64) ? (row0 + 4 + d) : (row0 + 68 + (d - 64));
        __builtin_amdgcn_global_load_async_to_lds_b32(
            (as1_int*)const_cast<float*>(g0),
            (as3_int*)&sbuf[0][d], 0, 0);
    }
#endif

    for (int l = 0; l < 64; ++l) {
#if SCAN_ASYNC
        if (l + 1 < 64) {
            const float* rown = xdbl + ((size_t)b * 64 + l + 1) * 144;
            const float* gn = (d < 64) ? (rown + 4 + d) : (rown + 68 + (d - 64));
            __builtin_amdgcn_global_load_async_to_lds_b32(
                (as1_int*)const_cast<float*>(gn),
                (as3_int*)&sbuf[(l + 1) & 1][d], 0, 0);
#if __has_builtin(__builtin_amdgcn_s_wait_asynccnt)
            __builtin_amdgcn_s_wait_asynccnt(1);   // step-l copy done (in-order)
#else
            asm volatile("s_wait_asynccnt 0x1" ::: "memory");
#endif
        } else {
#if __has_builtin(__builtin_amdgcn_s_wait_asynccnt)
            __builtin_amdgcn_s_wait_asynccnt(0);
#else
            asm volatile("s_wait_asynccnt 0x0" ::: "memory");
#endif
        }
        __syncthreads();
        const float* lbp = &sbuf[l & 1][0];
        const float* lcp = &sbuf[l & 1][64];
#else
        {
            const float* row = xdbl + ((size_t)b * 64 + l) * 144;
            if (d < 64) sbuf[0][d] = row[4 + d];
            else        sbuf[0][d] = row[68 + (d - 64)];
        }
        __syncthreads();
        const float* lbp = &sbuf[0][0];
        const float* lcp = &sbuf[0][64];
#endif
        float dtl = dt[((size_t)b * 64 + l) * 128 + d];
        float ul  = u[((size_t)b * 64 + l) * 128 + d];
        float dtu = dtl * ul;
        float accv = 0.f;
#pragma unroll
        for (int s = 0; s < 64; ++s) {
            float dA = __expf(dtl * Ar[s]);
            float hn = h[s] * dA + dtu * lbp[s];
            h[s] = hn;
            accv += hn * lcp[s];
        }
        y[((size_t)b * 64 + l) * 128 + d] = accv + ul * Dd;
        __syncthreads();   // protect sbuf[(l)&1] before it is overwritten at l+2
    }
}

// y *= silu(z)   (z = cols 128..255 of xz), in-place
__global__ void k_gate(const float* __restrict__ xz, float* __restrict__ y, int total) {
    int i = blockIdx.x * blockDim.x + threadIdx.x;
    if (i >= total) return;
    int d = i & 127;
    int r = i >> 7;
    float z = xz[(size_t)r * 256 + 128 + d];
    y[i] = y[i] * (z * sigmoidf_(z));
}

// ---------------------------------------------------------------------------
extern "C" void kernel_launch(void* const* d_in, const int* in_sizes, int n_in,
                              void* d_out, int out_size, void* d_ws, size_t ws_size,
                              hipStream_t stream) {
    (void)in_sizes; (void)n_in; (void)out_size; (void)ws_size;
    const float* x    = (const float*)d_in[0];
    const float* w1   = (const float*)d_in[1];  const float* b1  = (const float*)d_in[2];
    const float* w2   = (const float*)d_in[3];  const float* b2  = (const float*)d_in[4];
    const float* w3   = (const float*)d_in[5];  const float* b3  = (const float*)d_in[6];
    const float* wt1  = (const float*)d_in[7];  const float* bt1 = (const float*)d_in[8];
    const float* wt2  = (const float*)d_in[9];  const float* bt2 = (const float*)d_in[10];
    const float* wt3  = (const float*)d_in[11]; const float* bt3 = (const float*)d_in[12];
    const float* wt4  = (const float*)d_in[13]; const float* bt4 = (const float*)d_in[14];
    const float* g1   = (const float*)d_in[15]; const float* be1 = (const float*)d_in[16];
    const float* g2   = (const float*)d_in[17]; const float* be2 = (const float*)d_in[18];
    const float* g3   = (const float*)d_in[19]; const float* be3 = (const float*)d_in[20];
    const float* g4   = (const float*)d_in[21]; const float* be4 = (const float*)d_in[22];
    const float* g5   = (const float*)d_in[23]; const float* be5 = (const float*)d_in[24];
    const float* g6   = (const float*)d_in[25]; const float* be6 = (const float*)d_in[26];
    const float* win  = (const float*)d_in[27];
    const float* cw   = (const float*)d_in[28]; const float* cb  = (const float*)d_in[29];
    const float* wx   = (const float*)d_in[30];
    const float* wdt  = (const float*)d_in[31]; const float* bdt = (const float*)d_in[32];
    const float* Alog = (const float*)d_in[33]; const float* Dp  = (const float*)d_in[34];
    const float* wout = (const float*)d_in[35];
    float* out = (float*)d_out;

    const int B = 128;
    // -------- workspace carve-out (floats) --------
    float* ws = (float*)d_ws;
    size_t off = 0;
    auto alloc = [&](size_t n) { float* p = ws + off; off += n; return p; };
    float* x0pre = alloc(524288);  float* x0 = alloc(524288);   // (B,16,256)
    float* x1pre = alloc(524288);  float* x1 = alloc(524288);   // (B,32,128)
    float* x2pre = alloc(524288);  float* x2 = alloc(524288);   // (B,64,64)
    float* xz    = alloc(2097152);                              // (B*64,256)
    float* u     = alloc(1048576);                              // (B*64,128)
    float* xdbl  = alloc(1179648);                              // (B*64,144)
    float* dt    = alloc(1048576);                              // (B*64,128)
    float* ybuf  = alloc(1048576);                              // (B*64,128)
    float* x3    = alloc(524288);                               // (B*64,64)
    float* x5pre = alloc(524288);  float* x5 = alloc(524288);
    float* x6pre = alloc(524288);  float* x6 = alloc(524288);
    float* x7pre = alloc(524288);  float* x8 = alloc(524288);
    float* Abuf  = alloc(8192);                                 // -exp(Alog)
    float* wxp   = alloc(18432);                                // wx padded 144x128
    float* stats = alloc(128);                                  // per-BN scale/shift

    auto blk = [](int n) { return (n + 255) / 256; };

    // -------- prep --------
    k_prep<<<72, 256, 0, stream>>>(Alog, wx, Abuf, wxp);

    // -------- encoder --------
    k_conv_glu<<<blk(B * 16 * 256), 256, 0, stream>>>(x, w1, b1, x0pre, B, 1, 16, 512, 256);
    k_bn_stats<<<16, 256, 0, stream>>>(x0pre, g1, be1, stats, B, 16, 256);
    k_bn_apply<<<blk(B * 16 * 256), 256, 0, stream>>>(x0pre, stats, x0, 16, 256, B * 16 * 256);

    k_conv_glu<<<blk(B * 32 * 128), 256, 0, stream>>>(x0, w2, b2, x1pre, B, 16, 32, 256, 128);
    k_bn_stats<<<32, 256, 0, stream>>>(x1pre, g2, be2, stats, B, 32, 128);
    k_bn_apply<<<blk(B * 32 * 128), 256, 0, stream>>>(x1pre, stats, x1, 32, 128, B * 32 * 128);

    k_conv_glu<<<blk(B * 64 * 64), 256, 0, stream>>>(x1, w3, b3, x2pre, B, 32, 64, 128, 64);
    k_bn_stats<<<64, 256, 0, stream>>>(x2pre, g3, be3, stats, B, 64, 64);
    k_bn_apply<<<blk(B * 64 * 64), 256, 0, stream>>>(x2pre, stats, x2, 64, 64, B * 64 * 64);

    // -------- Mamba --------
    const int M = B * 64;  // 8192 rows
    // xz = x2 @ win^T : (8192,256,K=64)
    k_gemm_wmma_f32<<<(M / 16) * (256 / 16) / 8, 256, 0, stream>>>(
        x2, win, xz, M, 256, 64, 64, 64, 256, nullptr, 0);
    // depthwise causal conv + silu -> u
    k_dwconv_silu<<<blk(M * 128), 256, 0, stream>>>(xz, cw, cb, u, B);
    // x_dbl = u @ wxp^T : (8192,144,K=128)
    k_gemm_wmma_f32<<<(M / 16) * (144 / 16) / 8, 256, 0, stream>>>(
        u, wxp, xdbl, M, 144, 128, 128, 128, 144, nullptr, 0);
    // dt = softplus(dtp @ wdt^T + bdt) : (8192,128,K=4), A = xdbl cols 0..3
    k_gemm_wmma_f32<<<(M / 16) * (128 / 16) / 8, 256, 0, stream>>>(
        xdbl, wdt, dt, M, 128, 4, 144, 4, 128, bdt, 1);
    // selective scan (double-buffered async LDS staging)
    k_mamba_scan<<<B, 128, 0, stream>>>(dt, u, xdbl, Abuf, Dp, ybuf);
    // gate: y *= silu(z)
    k_gate<<<blk(M * 128), 256, 0, stream>>>(xz, ybuf, M * 128);
    // x3 = y @ wout^T : (8192,64,K=128)
    k_gemm_wmma_f32<<<(M / 16) * (64 / 16) / 8, 256, 0, stream>>>(
        ybuf, wout, x3, M, 64, 128, 128, 128, 64, nullptr, 0);

    // -------- decoder --------
    k_convT<<<blk(B * 64 * 64), 256, 0, stream>>>(x3, wt1, bt1, x2, x5pre,
                                                  B, 64, 64, 64, 64, 1, 1);
    k_bn_stats<<<64, 256, 0, stream>>>(x5pre, g4, be4, stats, B, 64, 64);
    k_bn_apply<<<blk(B * 64 * 64), 256, 0, stream>>>(x5pre, stats, x5, 64, 64, B * 64 * 64);

    k_convT<<<blk(B * 32 * 128), 256, 0, stream>>>(x5, wt2, bt2, x1, x6pre,
                                                   B, 64, 32, 64, 128, 2, 1);
    k_bn_stats<<<32, 256, 0, stream>>>(x6pre, g5, be5, stats, B, 32, 128);
    k_bn_apply<<<blk(B * 32 * 128), 256, 0, stream>>>(x6pre, stats, x6, 32, 128, B * 32 * 128);

    k_convT<<<blk(B * 16 * 256), 256, 0, stream>>>(x6, wt3, bt3, x0, x7pre,
                                                   B, 32, 16, 128, 256, 2, 1);
    k_bn_stats<<<16, 256, 0, stream>>>(x7pre, g6, be6, stats, B, 16, 256);
    k_bn_apply<<<blk(B * 16 * 256), 256, 0, stream>>>(x7pre, stats, x8, 16, 256, B * 16 * 256);

    k_convT<<<blk(B * 1 * 512), 256, 0, stream>>>(x8, wt4, bt4, nullptr, out,
                                                  B, 16, 1, 256, 512, 2, 0);
}